// MNESynapse_3633542332471
// MI455X (gfx1250) — compile-verified
//
#include <hip/hip_runtime.h>

// MNE synapse update, fused for MI455X (gfx1250, wave32).
// Memory-bound (~537MB/call); GEMMs done in exact f32 via V_WMMA_F32_16X16X4_F32.

typedef __attribute__((ext_vector_type(2))) float v2f;
typedef __attribute__((ext_vector_type(8))) float v8f;

#define MNE_N 4096
#define MNE_B 64

#define ETA 0.01f
#define MU 0.1f
#define GAMMA 0.05f
#define FORM_TH 0.5f
#define ELIM_TH 0.01f
#define NEW_W_SCALE 0.01f   // W_STD * 0.1

// --- Kernel 1: avg_pre[j] = mean_b pre[b, j] ---------------------------------
__global__ __launch_bounds__(256) void mne_avgpre_kernel(
    const float* __restrict__ pre, float* __restrict__ avg) {
  const int j = blockIdx.x * blockDim.x + threadIdx.x;
  if (j < MNE_N) {
    float s = 0.0f;
#pragma unroll 8
    for (int b = 0; b < MNE_B; ++b) s += pre[b * MNE_N + j];
    avg[j] = s * (1.0f / (float)MNE_B);
  }
}

// --- Kernel 2: fused dual-GEMM + elementwise update --------------------------
// Each wave owns one 16x16 output tile (ti, tj). 8 waves per block share ti
// so the A-operand loads (post/contrib columns) hit the WGP cache.
__global__ __launch_bounds__(256) void mne_update_kernel(
    const float* __restrict__ pre,      // [B,N]
    const float* __restrict__ post,     // [B,N]
    const float* __restrict__ contrib,  // [B,N]
    const float* __restrict__ weights,  // [N,N]
    const float* __restrict__ age,      // [N,N]
    const int*   __restrict__ conn,     // [N,N] (bool as int)
    const float* __restrict__ randw,    // [N,N]
    const float* __restrict__ avg_pre,  // [N]
    float* __restrict__ out_w,
    float* __restrict__ out_energy,
    float* __restrict__ out_age,
    float* __restrict__ out_mask) {
  const int lane = threadIdx.x & 31;
  const int wave = threadIdx.x >> 5;
  const int tj = blockIdx.x * 8 + wave;   // j tile
  const int ti = blockIdx.y;              // i tile
  const int half = lane >> 4;             // 0: K={0,1}, 1: K={2,3} (per VGPR pair)
  const int lm = lane & 15;

  const int i_a = ti * 16 + lm;           // A matrix M index (output row block)
  const int j_b = tj * 16 + lm;           // B matrix N index (output col block)

  v8f acc_heb = {};   // (post*contrib)^T x pre  accumulator
  v8f acc_cor = {};   //  post^T x pre           accumulator

  // K = B = 64 reduction: 16 steps of 16x16x4 f32 WMMA for each GEMM.
  // 32-bit A 16x4 layout: lanes 0-15 hold K=0 (v0) / K=1 (v1) at M=lane;
  //                       lanes 16-31 hold K=2 / K=3. B 4x16 mirrors this.
#pragma unroll
  for (int kk = 0; kk < 16; ++kk) {
    const int k0 = kk * 4 + half * 2;
    const float p0 = post[k0 * MNE_N + i_a];
    const float p1 = post[(k0 + 1) * MNE_N + i_a];
    const float c0 = contrib[k0 * MNE_N + i_a];
    const float c1 = contrib[(k0 + 1) * MNE_N + i_a];
    v2f b;
    b.x = pre[k0 * MNE_N + j_b];
    b.y = pre[(k0 + 1) * MNE_N + j_b];
    v2f a_h; a_h.x = p0 * c0; a_h.y = p1 * c1;
    v2f a_c; a_c.x = p0;      a_c.y = p1;
    acc_heb = __builtin_amdgcn_wmma_f32_16x16x4_f32(
        false, a_h, false, b, (short)0, acc_heb, false, false);
    acc_cor = __builtin_amdgcn_wmma_f32_16x16x4_f32(
        false, a_c, false, b, (short)0, acc_cor, false, false);
  }

  // C/D layout: VGPR r, lanes 0-15 -> (M=r, N=lane); lanes 16-31 -> (M=r+8).
  // Column j is lane-invariant across r, so avg_pre loads once per lane.
  const float apre = avg_pre[j_b];
  const float apre_abs = fabsf(apre);
  const float eta_b = ETA / (float)MNE_B;
  const float inv_b = 1.0f / (float)MNE_B;

#pragma unroll
  for (int r = 0; r < 8; ++r) {
    const int m = r + half * 8;
    const int idx = (ti * 16 + m) * MNE_N + j_b;

    const float w  = weights[idx];
    const float ag = age[idx];
    const bool  isconn = conn[idx] != 0;
    const float cf = isconn ? 1.0f : 0.0f;
    const float rw = randw[idx];

    const float heb = acc_heb[r] * eta_b;
    const float met = (-MU * GAMMA) * fabsf(w) * w * apre;
    float nw = w + (heb + met) * cf;
    nw = fminf(fmaxf(nw, -1.0f), 1.0f) * cf;

    const float energy = GAMMA * fabsf(w) * apre_abs;
    const float nage = ag + cf;

    const float corr = acc_cor[r] * inv_b;
    const bool newc = (corr > FORM_TH) && !isconn;
    if (newc) nw = rw * NEW_W_SCALE;
    bool maskb = isconn || newc;
    const bool weak = (fabsf(nw) < ELIM_TH) && maskb;
    if (weak) nw = 0.0f;
    maskb = maskb && !weak;

    out_w[idx]      = nw;
    out_energy[idx] = energy;
    out_age[idx]    = nage;
    out_mask[idx]   = maskb ? 1.0f : 0.0f;
  }
}

extern "C" void kernel_launch(void* const* d_in, const int* in_sizes, int n_in,
                              void* d_out, int out_size, void* d_ws, size_t ws_size,
                              hipStream_t stream) {
  // setup_inputs order:
  const float* pre     = (const float*)d_in[0];  // presynaptic  [B,N]
  const float* post    = (const float*)d_in[1];  // postsynaptic [B,N]
  const float* contrib = (const float*)d_in[2];  // contribution [B,N]
  const float* weights = (const float*)d_in[3];  // [N,N]
  const float* age     = (const float*)d_in[4];  // [N,N]
  const int*   conn    = (const int*)d_in[5];    // is_connected [N,N]
  const float* randw   = (const float*)d_in[6];  // new_weights_rand [N,N]

  float* out = (float*)d_out;                    // 4 * N*N floats, return order
  float* avg = (float*)d_ws;                     // N floats scratch

  mne_avgpre_kernel<<<MNE_N / 256, 256, 0, stream>>>(pre, avg);

  const size_t nn = (size_t)MNE_N * MNE_N;
  dim3 grid(MNE_N / 16 / 8, MNE_N / 16);         // 32 x 256 blocks, 8 waves each
  mne_update_kernel<<<grid, 256, 0, stream>>>(
      pre, post, contrib, weights, age, conn, randw, avg,
      out, out + nn, out + 2 * nn, out + 3 * nn);
}